// ResonantProjection_75892072121059
// MI455X (gfx1250) — compile-verified
//
#include <hip/hip_runtime.h>
#include <hip/hip_bf16.h>
#include <math.h>

#define LUT_SIZE 4096
#define IN_FEAT  1024
#define BATCH    64
#define NEURONS  2048
#define OUT_FEAT 1024
#define K_CH     128   // k-chunk staged in LDS per iteration

typedef __attribute__((ext_vector_type(2))) float v2f;
typedef __attribute__((ext_vector_type(8))) float v8f;

// ---------------------------------------------------------------------------
// Phase 1: interference[b, n] = sum_k comb[ idx(x[b,k]/(1+|W[n,k]|) + B[n,k]) ]
//   comb[i] = sin_table[i] + cos_table[i]  (fused once per block into LDS)
// Mapping: block = 4 neurons x 64 batches; 8 waves; wave w ->
//   neuron  = blockIdx.x*4 + (w>>1), batch half = w&1, lane = batch in half.
// W/B/rcp are wave-uniform; each lane owns one running sum (no reduction).
// ---------------------------------------------------------------------------
__global__ __launch_bounds__(256)
void resonant_interference_kernel(const float* __restrict__ x,
                                  const float* __restrict__ W,
                                  const float* __restrict__ Bm,
                                  const float* __restrict__ sin_t,
                                  const float* __restrict__ cos_t,
                                  float* __restrict__ inter) {
    __shared__ float comb[LUT_SIZE];        // 16 KB fused sin+cos LUT
    __shared__ float xs[K_CH * 65];         // x chunk, transposed, pad 65 -> no bank conflicts

    const int tid  = threadIdx.x;
    const int wave = tid >> 5;
    const int lane = tid & 31;

    // Build fused LUT (guarded by the first __syncthreads below)
    for (int i = tid; i < LUT_SIZE; i += 256)
        comb[i] = sin_t[i] + cos_t[i];

    const int n     = blockIdx.x * 4 + (wave >> 1);
    const int batch = (wave & 1) * 32 + lane;

    const float SCALE = (float)(LUT_SIZE / (2.0 * M_PI));  // 651.89865...
    float acc = 0.0f;

    for (int k0 = 0; k0 < IN_FEAT; k0 += K_CH) {
        __syncthreads();  // xs reuse fence (also fences comb on first pass)
        // Stage x[0:64, k0:k0+K_CH] transposed: xs[kl*65 + b]
        for (int f = tid; f < BATCH * K_CH; f += 256) {
            int b  = f >> 7;             // f / K_CH  (K_CH == 128)
            int kl = f & (K_CH - 1);
            xs[kl * 65 + b] = x[b * IN_FEAT + k0 + kl];
        }
        __syncthreads();

        const float* wrow = W  + (size_t)n * IN_FEAT + k0;
        const float* brow = Bm + (size_t)n * IN_FEAT + k0;
        #pragma unroll 4
        for (int kl = 0; kl < K_CH; ++kl) {
            float w  = wrow[kl];                    // wave-uniform -> 1 cacheline
            float bb = brow[kl];
            float rw = 1.0f / (1.0f + fabsf(w));    // wave-uniform v_rcp
            float theta = fmaf(xs[kl * 65 + batch], rw, bb);
            // jnp.mod(jnp.round(theta*SCALE), 4096): RNE round, & wraps negatives
            int idx = (int)rintf(theta * SCALE) & (LUT_SIZE - 1);
            acc += comb[idx];                       // 1 LDS gather per element
        }
    }
    inter[(size_t)batch * NEURONS + n] = acc;
}

// ---------------------------------------------------------------------------
// Phase 2: out(64x1024) = inter(64x2048) @ out_w^T   via V_WMMA_F32_16X16X4_F32
// One wave per 16x16 output tile: 4 M-tiles x 64 N-tiles = 256 waves.
// A 16x4 f32 layout: lanes 0-15 hold rows M=lane, K={0,1} in v[0],v[1];
//                    lanes 16-31 hold K={2,3}. B mirrors with N=lane&15.
// C/D: VGPR j holds M = j + 8*(lane>=16), N = lane&15.
// ---------------------------------------------------------------------------
__global__ __launch_bounds__(256)
void resonant_project_wmma_kernel(const float* __restrict__ inter,
                                  const float* __restrict__ out_w,
                                  float* __restrict__ out) {
    const int tid  = threadIdx.x;
    const int lane = tid & 31;
    const int gw   = blockIdx.x * 8 + (tid >> 5); // global wave id, 0..255
    const int mt   = gw & 3;                      // 4 tiles over batch
    const int nt   = gw >> 2;                     // 64 tiles over out_features
    const int half = lane >> 4;
    const int r    = lane & 15;

    // Per-lane fragment base pointers (8-byte aligned: even float offsets)
    const float* arow = inter + (size_t)(mt * 16 + r) * NEURONS + 2 * half;
    const float* brow = out_w + (size_t)(nt * 16 + r) * NEURONS + 2 * half;

    v8f acc = {};
    #pragma unroll 8
    for (int k0 = 0; k0 < NEURONS; k0 += 4) {
        v2f a = *(const v2f*)(arow + k0);   // A[M=r, K=k0+2*half .. +1]
        v2f b = *(const v2f*)(brow + k0);   // B[K, N=r] = out_w[N, K]
        // (neg_a, A, neg_b, B, c_mod, C, reuse_a, reuse_b)
        acc = __builtin_amdgcn_wmma_f32_16x16x4_f32(
            false, a, false, b, (short)0, acc, false, false);
    }

    float* orow = out + (size_t)(mt * 16) * OUT_FEAT + nt * 16 + r;
    #pragma unroll
    for (int j = 0; j < 8; ++j)
        orow[(size_t)(j + 8 * half) * OUT_FEAT] = acc[j];
}

// ---------------------------------------------------------------------------
extern "C" void kernel_launch(void* const* d_in, const int* in_sizes, int n_in,
                              void* d_out, int out_size, void* d_ws, size_t ws_size,
                              hipStream_t stream) {
    const float* x     = (const float*)d_in[0];  // (64, 1024)
    const float* W     = (const float*)d_in[1];  // (2048, 1024)
    const float* Bm    = (const float*)d_in[2];  // (2048, 1024)
    const float* out_w = (const float*)d_in[3];  // (1024, 2048)
    const float* sin_t = (const float*)d_in[4];  // (4096,)
    const float* cos_t = (const float*)d_in[5];  // (4096,)
    float* out   = (float*)d_out;                // (64, 1024)
    float* inter = (float*)d_ws;                 // (64, 2048) = 512 KB scratch

    // Phase 1: 512 blocks x (4 neurons x 64 batches)
    resonant_interference_kernel<<<NEURONS / 4, 256, 0, stream>>>(
        x, W, Bm, sin_t, cos_t, inter);

    // Phase 2: 256 WMMA tiles, 8 waves/block -> 32 blocks
    resonant_project_wmma_kernel<<<32, 256, 0, stream>>>(inter, out_w, out);
}